// BehavioralAttentionHead_20538533609916
// MI455X (gfx1250) — compile-verified
//
#include <hip/hip_runtime.h>
#include <math.h>

#define N_NODES 50000
#define N_EDGES 800000
#define FDIM    96
#define NH      8
#define HD      12
#define SLOPE   0.2f
#define LN_EPS  1e-5f

typedef float v2f __attribute__((ext_vector_type(2)));
typedef float v8f __attribute__((ext_vector_type(8)));

// Order-preserving float <-> uint encoding so atomicMax(uint) == float max.
__device__ __forceinline__ unsigned enc_ord(float f) {
    unsigned u = __float_as_uint(f);
    return (u & 0x80000000u) ? ~u : (u | 0x80000000u);
}
__device__ __forceinline__ float dec_ord(unsigned u) {
    return (u & 0x80000000u) ? __uint_as_float(u & 0x7FFFFFFFu)
                             : __uint_as_float(~u);
}

// -------- init: accum=0 (N*96), denom=0 (N*8), smax=enc(-inf) (N*8) --------
__global__ void init_kernel(float* __restrict__ accum,
                            float* __restrict__ denom,
                            unsigned* __restrict__ smax) {
    int i = blockIdx.x * blockDim.x + threadIdx.x;
    if (i < N_NODES * FDIM) accum[i] = 0.0f;
    if (i < N_NODES * NH) {
        denom[i] = 0.0f;
        smax[i]  = 0x007FFFFFu;   // enc_ord(-inf)
    }
}

// -------- fp32 WMMA GEMM: out = X @ W + b, one wave per 16x16 tile --------
// blockIdx.z selects (W_l,b_l,out=xl) or (W_r,b_r,out=xr).
__global__ void gemm_bias_wmma(const float* __restrict__ X,
                               const float* __restrict__ W0, const float* __restrict__ b0,
                               const float* __restrict__ W1, const float* __restrict__ b1,
                               float* __restrict__ out0, float* __restrict__ out1) {
    const float* W  = (blockIdx.z == 0) ? W0  : W1;
    const float* b  = (blockIdx.z == 0) ? b0  : b1;
    float*       out= (blockIdx.z == 0) ? out0: out1;

    const int m0   = blockIdx.x * 16;
    const int n0   = blockIdx.y * 16;
    const int lane = threadIdx.x;          // 0..31, one wave per block
    const int ll   = lane & 15;
    const int half = lane >> 4;
    const int koff = half << 1;            // 0 or 2 (A/B K-slot per ISA layout)

    // C/D layout: VGPR r -> M = r + half*8, N = ll. Bias depends only on N.
    v8f c;
    const float bv = b[n0 + ll];
#pragma unroll
    for (int r = 0; r < 8; ++r) c[r] = bv;

#pragma unroll
    for (int k = 0; k < FDIM; k += 4) {
        // A 16x4 f32: lane ll holds M=ll; VGPR0=K=koff, VGPR1=K=koff+1
        v2f a;
        a[0] = X[(m0 + ll) * FDIM + (k + koff)];
        a[1] = X[(m0 + ll) * FDIM + (k + koff + 1)];
        // B 4x16 f32: VGPR0 = row K=koff over N=ll; VGPR1 = row K=koff+1
        v2f bm;
        bm[0] = W[(k + koff)     * FDIM + n0 + ll];
        bm[1] = W[(k + koff + 1) * FDIM + n0 + ll];
        c = __builtin_amdgcn_wmma_f32_16x16x4_f32(
                /*neg_a=*/false, a, /*neg_b=*/false, bm,
                /*c_mod=*/(short)0, c, /*reuse_a=*/false, /*reuse_b=*/false);
    }

#pragma unroll
    for (int r = 0; r < 8; ++r)
        out[(m0 + r + half * 8) * FDIM + n0 + ll] = c[r];
}

// -------- per-edge GATv2 scoring + segment max (atomicMax on encoded f32) --
__global__ void edge_score_kernel(const float* __restrict__ xl,
                                  const float* __restrict__ xr,
                                  const int*   __restrict__ ei,
                                  const float* __restrict__ ew,
                                  const float* __restrict__ We,
                                  const float* __restrict__ att,
                                  float*    __restrict__ score,
                                  unsigned* __restrict__ smax) {
    int e = blockIdx.x * blockDim.x + threadIdx.x;
    if (e >= N_EDGES) return;
    const int src = ei[e];
    const int dst = ei[N_EDGES + e];
    const float w = ew[e];
    const float* xs = xl + (size_t)src * FDIM;
    const float* xd = xr + (size_t)dst * FDIM;

    float acc[NH];
#pragma unroll
    for (int h = 0; h < NH; ++h) acc[h] = 0.0f;

#pragma unroll
    for (int idx = 0; idx < FDIM; ++idx) {
        float v = xs[idx] + xd[idx] + w * We[idx];
        v = (v > 0.0f) ? v : SLOPE * v;          // leaky_relu(0.2)
        acc[idx / HD] += v * att[idx];           // att flat [H*D]
    }

#pragma unroll
    for (int h = 0; h < NH; ++h) {
        score[(size_t)e * NH + h] = acc[h];
        atomicMax(&smax[dst * NH + h], enc_ord(acc[h]));
    }
}

// -------- exp(score - smax[dst]) and segment-sum denominator ---------------
__global__ void edge_exp_kernel(const int* __restrict__ ei,
                                float* __restrict__ score,      // in: score, out: ex
                                const unsigned* __restrict__ smax,
                                float* __restrict__ denom) {
    int t = blockIdx.x * blockDim.x + threadIdx.x;
    if (t >= N_EDGES * NH) return;
    const int e = t >> 3;
    const int h = t & 7;
    const int dst = ei[N_EDGES + e];
    const float m  = dec_ord(smax[dst * NH + h]);
    const float ex = __expf(score[t] - m);
    score[t] = ex;
    atomicAdd(&denom[dst * NH + h], ex);
}

// -------- alpha-weighted scatter-add of xl[src] into accum[dst] ------------
__global__ void edge_scatter_kernel(const int* __restrict__ ei,
                                    const float* __restrict__ ex,
                                    const float* __restrict__ denom,
                                    const float* __restrict__ xl,
                                    float* __restrict__ accum) {
    int t = blockIdx.x * blockDim.x + threadIdx.x;
    if (t >= N_EDGES * NH) return;
    const int e = t >> 3;
    const int h = t & 7;
    const int src = ei[e];
    const int dst = ei[N_EDGES + e];
    const float alpha = ex[t] / (denom[dst * NH + h] + 1e-16f);
    const float* xs = xl + (size_t)src * FDIM + h * HD;
    float* ad = accum + (size_t)dst * FDIM + h * HD;
#pragma unroll
    for (int d = 0; d < HD; ++d)
        atomicAdd(&ad[d], alpha * xs[d]);
}

// -------- residual + LayerNorm, one wave32 per node (3 elems/lane) ---------
__global__ void ln_kernel(const float* __restrict__ accum,
                          const float* __restrict__ x,
                          const float* __restrict__ bias,
                          const float* __restrict__ gamma,
                          const float* __restrict__ beta,
                          float* __restrict__ out) {
    const int node = blockIdx.x * (blockDim.x >> 5) + (threadIdx.x >> 5);
    const int lane = threadIdx.x & 31;
    if (node >= N_NODES) return;

    float y[3];
    float sum = 0.0f;
#pragma unroll
    for (int i = 0; i < 3; ++i) {
        const int j = lane + 32 * i;
        y[i] = accum[(size_t)node * FDIM + j] + bias[j] + x[(size_t)node * FDIM + j];
        sum += y[i];
    }
#pragma unroll
    for (int m = 16; m > 0; m >>= 1) sum += __shfl_xor(sum, m, 32);
    const float mu = sum * (1.0f / FDIM);

    float vs = 0.0f;
#pragma unroll
    for (int i = 0; i < 3; ++i) { const float d = y[i] - mu; vs += d * d; }
#pragma unroll
    for (int m = 16; m > 0; m >>= 1) vs += __shfl_xor(vs, m, 32);
    const float inv = rsqrtf(vs * (1.0f / FDIM) + LN_EPS);

#pragma unroll
    for (int i = 0; i < 3; ++i) {
        const int j = lane + 32 * i;
        out[(size_t)node * FDIM + j] = (y[i] - mu) * inv * gamma[j] + beta[j];
    }
}

extern "C" void kernel_launch(void* const* d_in, const int* in_sizes, int n_in,
                              void* d_out, int out_size, void* d_ws, size_t ws_size,
                              hipStream_t stream) {
    const float* x     = (const float*)d_in[0];
    const int*   ei    = (const int*)  d_in[1];   // [2,E]
    const float* ew    = (const float*)d_in[2];   // [E,1]
    const float* W_l   = (const float*)d_in[3];
    const float* b_l   = (const float*)d_in[4];
    const float* W_r   = (const float*)d_in[5];
    const float* b_r   = (const float*)d_in[6];
    const float* W_e   = (const float*)d_in[7];   // [1,96]
    const float* att   = (const float*)d_in[8];   // [8,12]
    const float* bias  = (const float*)d_in[9];
    const float* gamma = (const float*)d_in[10];
    const float* beta  = (const float*)d_in[11];
    float* out = (float*)d_out;

    // workspace layout (floats): xl | xr | ex(E*8) | denom(N*8) | smax(N*8) | accum
    float* ws    = (float*)d_ws;
    float* xl    = ws;
    float* xr    = xl    + (size_t)N_NODES * FDIM;
    float* exbuf = xr    + (size_t)N_NODES * FDIM;
    float* denom = exbuf + (size_t)N_EDGES * NH;
    unsigned* smax = (unsigned*)(denom + (size_t)N_NODES * NH);
    float* accum = (float*)(smax + (size_t)N_NODES * NH);

    // 1) init accum/denom/smax
    {
        const int total = N_NODES * FDIM;
        init_kernel<<<(total + 255) / 256, 256, 0, stream>>>(accum, denom, smax);
    }
    // 2) xl = x@W_l+b_l ; xr = x@W_r+b_r  (fp32 WMMA, one wave per 16x16 tile)
    {
        dim3 grid(N_NODES / 16, FDIM / 16, 2);   // 50000 = 3125*16, 96 = 6*16
        gemm_bias_wmma<<<grid, 32, 0, stream>>>(x, W_l, b_l, W_r, b_r, xl, xr);
    }
    // 3) edge scores + segment max
    edge_score_kernel<<<(N_EDGES + 255) / 256, 256, 0, stream>>>(
        xl, xr, ei, ew, W_e, att, exbuf, smax);
    // 4) exp + segment sum
    {
        const int total = N_EDGES * NH;
        edge_exp_kernel<<<(total + 255) / 256, 256, 0, stream>>>(ei, exbuf, smax, denom);
    }
    // 5) weighted scatter-add
    {
        const int total = N_EDGES * NH;
        edge_scatter_kernel<<<(total + 255) / 256, 256, 0, stream>>>(
            ei, exbuf, denom, xl, accum);
    }
    // 6) bias + residual + LayerNorm
    ln_kernel<<<(N_NODES + 3) / 4, 128, 0, stream>>>(accum, x, bias, gamma, beta, out);
}